// PlainMamba_36206574305360
// MI455X (gfx1250) — compile-verified
//
#include <hip/hip_runtime.h>
#include <hip/hip_bf16.h>
#include <math.h>

// ---------------------------------------------------------------------------
// Mamba block for MI455X (gfx1250), fp32 end-to-end using V_WMMA_F32_16X16X4_F32
// with LDS double-buffered GEMM pipeline (B staged K-pair-interleaved so each
// B fragment is one aligned ds_load_b64 feeding the WMMA register pair).
// B=2, L=2048, D_MODEL=384, D_INNER=768, D_STATE=16, DT_RANK=24, K_CONV=4
// ---------------------------------------------------------------------------

#define B_SZ     2
#define L_SEQ    2048
#define D_MODEL  384
#define D_INNER  768
#define D_STATE  16
#define DT_RANK  24
#define K_CONV   4
#define ROWS     (B_SZ * L_SEQ)          // 4096 token rows

typedef float v2f __attribute__((ext_vector_type(2)));
typedef float v8f __attribute__((ext_vector_type(8)));

// ---------------------------------------------------------------------------
// fp32 WMMA GEMM with LDS double buffering.
// C[M,N] = A[M,K] * B[K,N] (row-major). Block tile 128x64, K staged by 16.
// Requires M%128==0, N%64==0, K%4==0 (K tail inside a stage is zero-padded).
// A tile in LDS: [row][16] row-major            -> A frag = ds_load_b64
// B tile in LDS: [kpair][col][2] interleaved    -> B frag = ds_load_b64
// ---------------------------------------------------------------------------
__global__ __launch_bounds__(256) void k_gemm_f32(
    const float* __restrict__ A, const float* __restrict__ B,
    float* __restrict__ C, int M, int N, int K, int lda, int ldb, int ldc)
{
    __shared__ float As[2][128 * 16];    // 2 x 8 KB
    __shared__ float Bs[2][8 * 128];     // 2 x 4 KB : [pair 0..7][col 0..63][2]

    const int tid  = threadIdx.x;
    const int lane = tid & 31;
    const int wave = tid >> 5;
    const int half = lane >> 4;          // K pair {0,1} vs {2,3}; C rows 0-7 vs 8-15
    const int r    = lane & 15;          // A row / B,C column within 16-wide tile

    const int m_blk = blockIdx.y * 128;
    const int n0g   = blockIdx.x * 64;

    // A staging: 128x16 tile, thread handles float4 f = tid and tid+256
    const int ar0 = tid >> 2;               // 4 float4 per row
    const int ac0 = (tid & 3) * 4;
    const int ar1 = (tid + 256) >> 2;
    const int ac1 = ((tid + 256) & 3) * 4;
    // B staging: 16x64 tile as 8 K-pairs x 64 cols; thread = (pair, 2 cols)
    const int bp  = tid >> 5;               // pair index 0..7  (rows 2p, 2p+1)
    const int bc  = (lane) * 2;             // columns bc, bc+1

    const int S = (K + 15) / 16;

    float4 ra0, ra1, rb;                    // in-flight stage registers

    auto gload = [&](int s, float4& a0, float4& a1, float4& b0) {
        const int k0 = s * 16;
        a0 = (k0 + ac0 < K) ? *(const float4*)&A[(size_t)(m_blk + ar0) * lda + k0 + ac0]
                            : make_float4(0.f, 0.f, 0.f, 0.f);
        a1 = (k0 + ac1 < K) ? *(const float4*)&A[(size_t)(m_blk + ar1) * lda + k0 + ac1]
                            : make_float4(0.f, 0.f, 0.f, 0.f);
        const int krow = k0 + 2 * bp;       // even; K%4==0 => krow<K implies krow+1<K
        if (krow < K) {
            float2 lo = *(const float2*)&B[(size_t)krow       * ldb + n0g + bc];
            float2 hi = *(const float2*)&B[(size_t)(krow + 1) * ldb + n0g + bc];
            b0 = make_float4(lo.x, hi.x, lo.y, hi.y);   // [col][2] interleave
        } else {
            b0 = make_float4(0.f, 0.f, 0.f, 0.f);
        }
    };
    auto sts = [&](int buf, const float4& a0, const float4& a1, const float4& b0) {
        *(float4*)&As[buf][ar0 * 16 + ac0]   = a0;
        *(float4*)&As[buf][ar1 * 16 + ac1]   = a1;
        *(float4*)&Bs[buf][bp * 128 + bc * 2] = b0;     // ds_store_b128
    };

    v8f acc[4] = {};

    gload(0, ra0, ra1, rb);
    sts(0, ra0, ra1, rb);

    for (int s = 0; s < S; ++s) {
        __syncthreads();                              // buf[s&1] visible to all
        if (s + 1 < S) gload(s + 1, ra0, ra1, rb);    // loads in flight over compute
        if (s + 2 < S) {                              // distance-2 cache prefetch
            const int k2 = (s + 2) * 16;
            __builtin_prefetch(&A[(size_t)(m_blk + ar0) * lda + k2 + ac0], 0, 0);
            __builtin_prefetch(&B[(size_t)(k2 + 2 * bp) * ldb + n0g + bc], 0, 0);
        }

        const float* Ab = &As[s & 1][(wave * 16 + r) * 16];
        const float* Bb = &Bs[s & 1][0];
#pragma unroll
        for (int k = 0; k < 16; k += 4) {
            v2f a = *(const v2f*)&Ab[k + 2 * half];               // ds_load_b64
            const int kp = (k + 2 * half) >> 1;                   // pair index
#pragma unroll
            for (int t = 0; t < 4; ++t) {
                v2f bb = *(const v2f*)&Bb[kp * 128 + (16 * t + r) * 2]; // ds_load_b64
                acc[t] = __builtin_amdgcn_wmma_f32_16x16x4_f32(
                    false, a, false, bb, (short)0, acc[t], false, false);
            }
        }
        if (s + 1 < S) sts((s + 1) & 1, ra0, ra1, rb);  // waits loads, after 16 WMMAs
    }

    const int m0 = m_blk + wave * 16;
#pragma unroll
    for (int t = 0; t < 4; ++t)
#pragma unroll
        for (int i = 0; i < 8; ++i)
            C[(size_t)(m0 + i + 8 * half) * ldc + n0g + 16 * t + r] = acc[t][i];
}

// ---------------------------------------------------------------------------
// Pad x_proj_w [768,56] -> [768,64] (zero-filled tail) so WMMA N-tiling works.
// ---------------------------------------------------------------------------
__global__ void k_pad_xprojw(const float* __restrict__ w, float* __restrict__ wp)
{
    int e = blockIdx.x * 256 + threadIdx.x;
    if (e >= D_INNER * 64) return;
    int row = e >> 6, col = e & 63;
    wp[e] = (col < (DT_RANK + 2 * D_STATE)) ? w[row * (DT_RANK + 2 * D_STATE) + col] : 0.0f;
}

// ---------------------------------------------------------------------------
// Depthwise causal conv (k=4) + bias + SiLU.  xp lives in xz[:, 0:768].
// ---------------------------------------------------------------------------
__global__ void k_conv_silu(const float* __restrict__ xz,
                            const float* __restrict__ conv_w,
                            const float* __restrict__ conv_b,
                            float* __restrict__ xpc)
{
    int e = blockIdx.x * 256 + threadIdx.x;
    if (e >= ROWS * D_INNER) return;
    int d  = e % D_INNER;
    int bl = e / D_INNER;
    int l  = bl % L_SEQ;
    int b  = bl / L_SEQ;

    float acc = conv_b[d];
#pragma unroll
    for (int j = 0; j < K_CONV; ++j) {
        int ls = l - (K_CONV - 1) + j;
        if (ls >= 0)
            acc += xz[((size_t)(b * L_SEQ + ls)) * (2 * D_INNER) + d] * conv_w[d * K_CONV + j];
    }
    xpc[e] = acc / (1.0f + __expf(-acc));     // silu
}

// ---------------------------------------------------------------------------
// dt = softplus(dt_pre + dt_proj_b)   (in place on the GEMM output)
// ---------------------------------------------------------------------------
__global__ void k_softplus_bias(float* __restrict__ dt, const float* __restrict__ bias)
{
    int e = blockIdx.x * 256 + threadIdx.x;
    if (e >= ROWS * D_INNER) return;
    float v = dt[e] + bias[e % D_INNER];
    dt[e] = (v > 20.0f) ? v : log1pf(__expf(v));
}

// ---------------------------------------------------------------------------
// Selective scan. One thread per (batch, channel); 16 h-states in registers.
// B/C vectors (cols 24..55 of padded x_dbl) staged in LDS in 64-step chunks.
// Also applies +xp*D and *silu(z) so output feeds out_proj directly.
// ---------------------------------------------------------------------------
__global__ __launch_bounds__(256) void k_scan(
    const float* __restrict__ xz,      // for z = xz[:, 768+d]
    const float* __restrict__ xpc,     // silu(conv(xp))
    const float* __restrict__ xdbl,    // padded [ROWS,64]: dt_r|B|C
    const float* __restrict__ dt,      // softplus'd
    const float* __restrict__ A_log,
    const float* __restrict__ Dvec,
    float* __restrict__ yg)
{
    const int b = blockIdx.x / (D_INNER / 256);
    const int d = (blockIdx.x % (D_INNER / 256)) * 256 + threadIdx.x;

    float An[D_STATE];
#pragma unroll
    for (int n = 0; n < D_STATE; ++n) An[n] = -__expf(A_log[d * D_STATE + n]);
    const float Dd = Dvec[d];

    float h[D_STATE];
#pragma unroll
    for (int n = 0; n < D_STATE; ++n) h[n] = 0.0f;

    __shared__ float sBC[64 * 32];     // 64 timesteps x (16 B + 16 C) = 8 KB

    for (int l0 = 0; l0 < L_SEQ; l0 += 64) {
        // cooperative stage of B/C chunk into LDS
        for (int idx = threadIdx.x; idx < 64 * 32; idx += 256) {
            int row = idx >> 5, col = idx & 31;
            sBC[idx] = xdbl[((size_t)(b * L_SEQ + l0 + row)) * 64 + DT_RANK + col];
        }
        __syncthreads();

        for (int i = 0; i < 64; ++i) {
            const size_t rb = (size_t)(b * L_SEQ + l0 + i);
            const float dtv = dt [rb * D_INNER + d];
            const float xpv = xpc[rb * D_INNER + d];
            const float zv  = xz [rb * (2 * D_INNER) + D_INNER + d];
            const float dtx = dtv * xpv;
            const float* Brow = &sBC[i * 32];

            float y = 0.0f;
#pragma unroll
            for (int n = 0; n < D_STATE; ++n) {
                float dA = __expf(dtv * An[n]);          // v_exp_f32 (TRANS, co-exec)
                h[n] = dA * h[n] + Brow[n] * dtx;
                y   += h[n] * Brow[16 + n];
            }
            y += xpv * Dd;
            float sz = zv / (1.0f + __expf(-zv));        // silu(z)
            yg[rb * D_INNER + d] = y * sz;
        }
        __syncthreads();
    }
}

// ---------------------------------------------------------------------------
extern "C" void kernel_launch(void* const* d_in, const int* in_sizes, int n_in,
                              void* d_out, int out_size, void* d_ws, size_t ws_size,
                              hipStream_t stream)
{
    const float* x         = (const float*)d_in[0];
    const float* in_proj_w = (const float*)d_in[1];
    const float* conv_w    = (const float*)d_in[2];
    const float* conv_b    = (const float*)d_in[3];
    const float* x_proj_w  = (const float*)d_in[4];
    const float* dt_proj_w = (const float*)d_in[5];
    const float* dt_proj_b = (const float*)d_in[6];
    const float* A_log     = (const float*)d_in[7];
    const float* Dvec      = (const float*)d_in[8];
    const float* out_proj_w= (const float*)d_in[9];
    float* out = (float*)d_out;

    // workspace layout (floats)
    float* ws   = (float*)d_ws;
    float* xz   = ws;                                   // 4096*1536
    float* xpc  = xz   + (size_t)ROWS * 2 * D_INNER;    // 4096*768
    float* xdbl = xpc  + (size_t)ROWS * D_INNER;        // 4096*64
    float* dt   = xdbl + (size_t)ROWS * 64;             // 4096*768
    float* yg   = dt   + (size_t)ROWS * D_INNER;        // 4096*768
    float* xpw  = yg   + (size_t)ROWS * D_INNER;        // 768*64

    // 1) pad x_proj_w -> [768,64]
    k_pad_xprojw<<<(D_INNER * 64 + 255) / 256, 256, 0, stream>>>(x_proj_w, xpw);

    // 2) xz = x @ in_proj_w        [4096,384]x[384,1536]
    k_gemm_f32<<<dim3((2 * D_INNER) / 64, ROWS / 128), 256, 0, stream>>>(
        x, in_proj_w, xz, ROWS, 2 * D_INNER, D_MODEL, D_MODEL, 2 * D_INNER, 2 * D_INNER);

    // 3) xp = silu(conv(xp) + b)
    k_conv_silu<<<(ROWS * D_INNER + 255) / 256, 256, 0, stream>>>(xz, conv_w, conv_b, xpc);

    // 4) x_dbl = xp @ x_proj_w_pad [4096,768]x[768,64]
    k_gemm_f32<<<dim3(64 / 64, ROWS / 128), 256, 0, stream>>>(
        xpc, xpw, xdbl, ROWS, 64, D_INNER, D_INNER, 64, 64);

    // 5) dt_pre = dt_r @ dt_proj_w [4096,24]x[24,768]  (dt_r = x_dbl[:, :24], lda=64)
    k_gemm_f32<<<dim3(D_INNER / 64, ROWS / 128), 256, 0, stream>>>(
        xdbl, dt_proj_w, dt, ROWS, D_INNER, DT_RANK, 64, D_INNER, D_INNER);

    // 6) dt = softplus(dt_pre + dt_proj_b)
    k_softplus_bias<<<(ROWS * D_INNER + 255) / 256, 256, 0, stream>>>(dt, dt_proj_b);

    // 7) selective scan + skip + gating -> yg
    k_scan<<<B_SZ * (D_INNER / 256), 256, 0, stream>>>(xz, xpc, xdbl, dt, A_log, Dvec, yg);

    // 8) out = yg @ out_proj_w     [4096,768]x[768,384]
    k_gemm_f32<<<dim3(D_MODEL / 64, ROWS / 128), 256, 0, stream>>>(
        yg, out_proj_w, out, ROWS, D_MODEL, D_INNER, D_INNER, D_MODEL, D_MODEL);
}